// GATLayer_1_41540923687148
// MI455X (gfx1250) — compile-verified
//
#include <hip/hip_runtime.h>
#include <math.h>

#define Nn 10000
#define Ee 320000
#define IN_CH 128
#define HID 64
#define HEADS 4
#define OUT_CH 64
#define F1 (HEADS*HID)   // 256
#define NEG_SLOPE 0.2f

typedef __attribute__((ext_vector_type(2))) float v2f;
typedef __attribute__((ext_vector_type(8))) float v8f;

// ---------------- utility kernels ----------------
__global__ void fill_kernel(float* p, float v, long long n) {
  long long i = blockIdx.x * 256LL + threadIdx.x;
  if (i < n) p[i] = v;
}

// ce1[h] = sum_c We1[h*64+c]*a_edge1[h*64+c] ; ce2[0] = sum_c We2[c]*a_edge2[c]
__global__ void compute_ce_kernel(const float* __restrict__ We1, const float* __restrict__ ae1,
                                  const float* __restrict__ We2, const float* __restrict__ ae2,
                                  float* ce1, float* ce2) {
  int t = threadIdx.x;
  if (t < HEADS) {
    float s = 0.f;
    for (int c = 0; c < HID; ++c) s += We1[t*HID+c] * ae1[t*HID+c];
    ce1[t] = s;
  } else if (t == HEADS) {
    float s = 0.f;
    for (int c = 0; c < OUT_CH; ++c) s += We2[c] * ae2[c];
    ce2[0] = s;
  }
}

// ---------------- WMMA fp32 GEMM: C[M,N] = A[M,K] @ B[K,N] ----------------
// One wave computes one 16x16 tile via V_WMMA_F32_16X16X4_F32, stepping K by 4.
// A 16x4 layout : lane&15 = M row, VGPR0/1 = K pair, lanes>=16 hold K=2,3.
// B 4x16 layout : lane&15 = N col, mirrored K mapping.
// C 16x16 layout: VGPR r -> M = r + (lane>>4)*8, N = lane&15.
__global__ void gemm_wmma_f32(const float* __restrict__ A, const float* __restrict__ B,
                              float* __restrict__ C, int M, int N, int K) {
  int wave = (blockIdx.x * blockDim.x + threadIdx.x) >> 5;
  int lane = threadIdx.x & 31;
  int tilesN = N >> 4;
  int tilesM = M >> 4;
  if (wave >= tilesM * tilesN) return;      // wave-uniform guard: EXEC all-ones for WMMA
  int tm = (wave / tilesN) << 4;
  int tn = (wave % tilesN) << 4;
  int r    = lane & 15;
  int half = lane >> 4;        // 0 or 1
  int kb   = half * 2;         // K offset within the group of 4

  v8f acc = {};
  const float* Arow = A + (size_t)(tm + r) * K;
  for (int k0 = 0; k0 < K; k0 += 4) {
    v2f a, b;
    a.x = Arow[k0 + kb];
    a.y = Arow[k0 + kb + 1];
    b.x = B[(size_t)(k0 + kb)     * N + tn + r];
    b.y = B[(size_t)(k0 + kb + 1) * N + tn + r];
    acc = __builtin_amdgcn_wmma_f32_16x16x4_f32(false, a, false, b,
                                                (short)0, acc, false, false);
  }
  float* Crow = C + (size_t)(tm + half * 8) * N + tn + r;
#pragma unroll
  for (int i = 0; i < 8; ++i) Crow[(size_t)i * N] = acc[i];
}

// ---------------- per-node attention logits ----------------
__global__ void node_logits_kernel(const float* __restrict__ xh,
                                   const float* __restrict__ a_src,
                                   const float* __restrict__ a_dst,
                                   float* al_s, float* al_d,
                                   int n, int heads, int hid) {
  int idx = blockIdx.x * blockDim.x + threadIdx.x;
  if (idx >= n * heads) return;
  int node = idx / heads, h = idx % heads;
  const float* v = xh + (size_t)node * heads * hid + (size_t)h * hid;
  float s = 0.f, d = 0.f;
  for (int c = 0; c < hid; ++c) {
    float x = v[c];
    s += x * a_src[h*hid + c];
    d += x * a_dst[h*hid + c];
  }
  al_s[idx] = s;
  al_d[idx] = d;
}

// ---------------- edge passes ----------------
__device__ inline void atomicMaxF(float* addr, float val) {
  if (val >= 0.0f) atomicMax((int*)addr, __float_as_int(val));
  else             atomicMin((unsigned int*)addr, __float_as_uint(val));
}

__device__ inline float lrelu(float a) { return (a > 0.f) ? a : NEG_SLOPE * a; }

// H=4 vectorized: one thread per edge, float4 across heads
__global__ void edge_alpha4_kernel(const int* __restrict__ src, const int* __restrict__ dst,
                                   const float* __restrict__ w,
                                   const float* __restrict__ al_s, const float* __restrict__ al_d,
                                   const float* __restrict__ ce,
                                   float* alpha, float* amax, int ne) {
  int e = blockIdx.x * blockDim.x + threadIdx.x;
  if (e >= ne) return;
  int s = src[e], d = dst[e];
  float4 as = *(const float4*)(al_s + (size_t)s * 4);
  float4 ad = *(const float4*)(al_d + (size_t)d * 4);
  float4 c  = *(const float4*)ce;
  float we  = w[e];
  float4 a;
  a.x = lrelu(as.x + ad.x + we * c.x);
  a.y = lrelu(as.y + ad.y + we * c.y);
  a.z = lrelu(as.z + ad.z + we * c.z);
  a.w = lrelu(as.w + ad.w + we * c.w);
  *(float4*)(alpha + (size_t)e * 4) = a;
  float* mp = amax + (size_t)d * 4;
  atomicMaxF(mp + 0, a.x);
  atomicMaxF(mp + 1, a.y);
  atomicMaxF(mp + 2, a.z);
  atomicMaxF(mp + 3, a.w);
}

__global__ void edge_exp4_kernel(const int* __restrict__ dst,
                                 float* alpha, const float* __restrict__ amax,
                                 float* denom, int ne) {
  int e = blockIdx.x * blockDim.x + threadIdx.x;
  if (e >= ne) return;
  int d = dst[e];
  float4 a = *(const float4*)(alpha + (size_t)e * 4);
  float4 m = *(const float4*)(amax + (size_t)d * 4);
  float4 ex;
  ex.x = expf(a.x - m.x);
  ex.y = expf(a.y - m.y);
  ex.z = expf(a.z - m.z);
  ex.w = expf(a.w - m.w);
  *(float4*)(alpha + (size_t)e * 4) = ex;
  float* dp = denom + (size_t)d * 4;
  atomicAdd(dp + 0, ex.x);
  atomicAdd(dp + 1, ex.y);
  atomicAdd(dp + 2, ex.z);
  atomicAdd(dp + 3, ex.w);
}

// generic scalar versions (used for H=1 layer)
__global__ void edge_alpha_kernel(const int* __restrict__ src, const int* __restrict__ dst,
                                  const float* __restrict__ w,
                                  const float* __restrict__ al_s, const float* __restrict__ al_d,
                                  const float* __restrict__ ce,
                                  float* alpha, float* amax, int ne, int heads) {
  int idx = blockIdx.x * blockDim.x + threadIdx.x;
  if (idx >= ne * heads) return;
  int e = idx / heads, h = idx % heads;
  int s = src[e], d = dst[e];
  float a = lrelu(al_s[s*heads + h] + al_d[d*heads + h] + w[e] * ce[h]);
  alpha[idx] = a;
  atomicMaxF(&amax[d*heads + h], a);
}

__global__ void edge_exp_kernel(const int* __restrict__ dst,
                                float* alpha, const float* __restrict__ amax,
                                float* denom, int ne, int heads) {
  int idx = blockIdx.x * blockDim.x + threadIdx.x;
  if (idx >= ne * heads) return;
  int e = idx / heads, h = idx % heads;
  int d = dst[e];
  float ex = expf(alpha[idx] - amax[d*heads + h]);
  alpha[idx] = ex;
  atomicAdd(&denom[d*heads + h], ex);
}

// aggregation: one thread per (edge, 4-channel chunk) -> b128 gather + 4 atomic adds
__global__ void edge_aggr4_kernel(const int* __restrict__ src, const int* __restrict__ dst,
                                  const float* __restrict__ ex, const float* __restrict__ denom,
                                  const float* __restrict__ xh, float* acc,
                                  int ne, int heads, int hid) {
  int F  = heads * hid;
  int F4 = F >> 2;
  long long idx = blockIdx.x * 256LL + threadIdx.x;
  if (idx >= (long long)ne * F4) return;
  int e = (int)(idx / F4);
  int q = (int)(idx % F4);          // 4-channel chunk within the row
  int h = (q << 2) / hid;
  int s = src[e], d = dst[e];
  float att = ex[e*heads + h] / (denom[d*heads + h] + 1e-16f);
  float4 v = ((const float4*)(xh + (size_t)s * F))[q];
  float* ap = acc + (size_t)d * F + (q << 2);
  atomicAdd(ap + 0, v.x * att);
  atomicAdd(ap + 1, v.y * att);
  atomicAdd(ap + 2, v.z * att);
  atomicAdd(ap + 3, v.w * att);
}

// ---------------- elementwise finalize ----------------
__global__ void bias_elu_kernel(const float* __restrict__ acc, const float* __restrict__ b,
                                float* out, int total, int F) {
  int i = blockIdx.x * blockDim.x + threadIdx.x;
  if (i >= total) return;
  float v = acc[i] + b[i % F];
  out[i] = (v > 0.f) ? v : expm1f(v);           // ELU (alpha=1)
}

__global__ void bias_add_kernel(float* out, const float* __restrict__ b, int total, int F) {
  int i = blockIdx.x * blockDim.x + threadIdx.x;
  if (i >= total) return;
  out[i] += b[i % F];
}

// ---------------- launch ----------------
extern "C" void kernel_launch(void* const* d_in, const int* in_sizes, int n_in,
                              void* d_out, int out_size, void* d_ws, size_t ws_size,
                              hipStream_t stream) {
  const float* x       = (const float*)d_in[0];
  const float* ew      = (const float*)d_in[1];
  const float* W1      = (const float*)d_in[2];
  const float* a_src1  = (const float*)d_in[3];
  const float* a_dst1  = (const float*)d_in[4];
  const float* a_edge1 = (const float*)d_in[5];
  const float* We1     = (const float*)d_in[6];
  const float* b1      = (const float*)d_in[7];
  const float* W2      = (const float*)d_in[8];
  const float* a_src2  = (const float*)d_in[9];
  const float* a_dst2  = (const float*)d_in[10];
  const float* a_edge2 = (const float*)d_in[11];
  const float* We2     = (const float*)d_in[12];
  const float* b2      = (const float*)d_in[13];
  const int*   eidx    = (const int*)d_in[14];
  const int* src = eidx;
  const int* dst = eidx + Ee;
  float* outf = (float*)d_out;

  float* ws = (float*)d_ws;
  float* xh1   = ws; ws += (size_t)Nn * F1;
  float* als1  = ws; ws += (size_t)Nn * HEADS;
  float* ald1  = ws; ws += (size_t)Nn * HEADS;
  float* amax1 = ws; ws += (size_t)Nn * HEADS;
  float* den1  = ws; ws += (size_t)Nn * HEADS;
  float* alp1  = ws; ws += (size_t)Ee * HEADS;
  float* acc1  = ws; ws += (size_t)Nn * F1;
  float* h1    = ws; ws += (size_t)Nn * F1;
  float* xh2   = ws; ws += (size_t)Nn * OUT_CH;
  float* als2  = ws; ws += Nn;
  float* ald2  = ws; ws += Nn;
  float* amax2 = ws; ws += Nn;
  float* den2  = ws; ws += Nn;
  float* alp2  = ws; ws += Ee;
  float* ce1   = ws; ws += 4;
  float* ce2   = ws; ws += 4;

  const int TB = 256;
  auto cdiv = [](long long a, long long b) { return (int)((a + b - 1) / b); };

  // precompute edge-attn scalars
  compute_ce_kernel<<<1, 32, 0, stream>>>(We1, a_edge1, We2, a_edge2, ce1, ce2);

  // ---- layer 1 ----
  {
    int waves = (Nn/16) * (F1/16);                 // 10000 waves
    gemm_wmma_f32<<<cdiv(waves*32, TB), TB, 0, stream>>>(x, W1, xh1, Nn, F1, IN_CH);
  }
  node_logits_kernel<<<cdiv(Nn*HEADS, TB), TB, 0, stream>>>(xh1, a_src1, a_dst1, als1, ald1,
                                                            Nn, HEADS, HID);
  fill_kernel<<<cdiv(Nn*HEADS, TB), TB, 0, stream>>>(amax1, -INFINITY, Nn*HEADS);
  fill_kernel<<<cdiv(Nn*HEADS, TB), TB, 0, stream>>>(den1, 0.0f, Nn*HEADS);
  fill_kernel<<<cdiv((long long)Nn*F1, TB), TB, 0, stream>>>(acc1, 0.0f, (long long)Nn*F1);

  edge_alpha4_kernel<<<cdiv(Ee, TB), TB, 0, stream>>>(src, dst, ew, als1, ald1, ce1,
                                                      alp1, amax1, Ee);
  edge_exp4_kernel<<<cdiv(Ee, TB), TB, 0, stream>>>(dst, alp1, amax1, den1, Ee);
  edge_aggr4_kernel<<<cdiv((long long)Ee*(F1/4), TB), TB, 0, stream>>>(
      src, dst, alp1, den1, xh1, acc1, Ee, HEADS, HID);
  bias_elu_kernel<<<cdiv((long long)Nn*F1, TB), TB, 0, stream>>>(acc1, b1, h1, Nn*F1, F1);

  // ---- layer 2 ----
  {
    int waves = (Nn/16) * (OUT_CH/16);             // 2500 waves
    gemm_wmma_f32<<<cdiv(waves*32, TB), TB, 0, stream>>>(h1, W2, xh2, Nn, OUT_CH, F1);
  }
  node_logits_kernel<<<cdiv(Nn, TB), TB, 0, stream>>>(xh2, a_src2, a_dst2, als2, ald2,
                                                      Nn, 1, OUT_CH);
  fill_kernel<<<cdiv(Nn, TB), TB, 0, stream>>>(amax2, -INFINITY, Nn);
  fill_kernel<<<cdiv(Nn, TB), TB, 0, stream>>>(den2, 0.0f, Nn);
  fill_kernel<<<cdiv((long long)Nn*OUT_CH, TB), TB, 0, stream>>>(outf, 0.0f, (long long)Nn*OUT_CH);

  edge_alpha_kernel<<<cdiv(Ee, TB), TB, 0, stream>>>(
      src, dst, ew, als2, ald2, ce2, alp2, amax2, Ee, 1);
  edge_exp_kernel<<<cdiv(Ee, TB), TB, 0, stream>>>(dst, alp2, amax2, den2, Ee, 1);
  edge_aggr4_kernel<<<cdiv((long long)Ee*(OUT_CH/4), TB), TB, 0, stream>>>(
      src, dst, alp2, den2, xh2, outf, Ee, 1, OUT_CH);
  bias_add_kernel<<<cdiv((long long)Nn*OUT_CH, TB), TB, 0, stream>>>(outf, b2, Nn*OUT_CH, OUT_CH);
}